// GAT_classifier_45079976739075
// MI455X (gfx1250) — compile-verified
//
#include <hip/hip_runtime.h>
#include <math.h>

#define N_NODES   100000
#define N_EDGES   1600000
#define NUM_GRAPHS 64
#define DIN  128
#define DLAT 64
#define NHEAD 2
#define NCLS 10

typedef __attribute__((ext_vector_type(2))) float v2f;
typedef __attribute__((ext_vector_type(8))) float v8f;

// ---- workspace layout (in floats) ----
#define OFF_H0    ((size_t)0)                         // N*64
#define OFF_AI    (OFF_H0   + (size_t)N_NODES*DLAT)   // N*2
#define OFF_AJ    (OFF_AI   + (size_t)N_NODES*NHEAD)  // N*2
#define OFF_EMAX  (OFF_AJ   + (size_t)N_NODES*NHEAD)  // N*2
#define OFF_DEN   (OFF_EMAX + (size_t)N_NODES*NHEAD)  // N*2
#define OFF_EE    (OFF_DEN  + (size_t)N_NODES*NHEAD)  // E*2
#define OFF_HAGG  (OFF_EE   + (size_t)N_EDGES*NHEAD)  // N*64
#define OFF_GSUM  (OFF_HAGG + (size_t)N_NODES*DLAT)   // 64*64
#define OFF_CNT   (OFF_GSUM + (size_t)NUM_GRAPHS*DLAT)// 64
#define WS_TOTAL  (OFF_CNT  + (size_t)NUM_GRAPHS)

// -------- init: zero denom/hagg/gsum/cnt, set emax = -inf --------
__global__ void gat_init_kernel(float* ws) {
    size_t stride = (size_t)gridDim.x * blockDim.x;
    size_t i0 = (size_t)blockIdx.x * blockDim.x + threadIdx.x;
    for (size_t i = i0; i < (size_t)N_NODES * NHEAD; i += stride) {
        ws[OFF_DEN  + i] = 0.0f;
        ws[OFF_EMAX + i] = -INFINITY;
    }
    for (size_t i = i0; i < (size_t)N_NODES * DLAT; i += stride)
        ws[OFF_HAGG + i] = 0.0f;
    for (size_t i = i0; i < (size_t)NUM_GRAPHS * (DLAT + 1); i += stride)
        ws[OFF_GSUM + i] = 0.0f;   // gsum then cnt are contiguous
}

// -------- h0 = x @ W0 + b0 via V_WMMA_F32_16X16X4_F32 --------
// one wave -> 16 rows x 64 cols; k-loop over 128 in steps of 4
__global__ void gat_gemm_h0_kernel(const float* __restrict__ x,
                                   const float* __restrict__ W0,
                                   const float* __restrict__ b0,
                                   float* __restrict__ h0) {
    int wave = (blockIdx.x * blockDim.x + threadIdx.x) >> 5;
    int lane = threadIdx.x & 31;
    if (wave >= N_NODES / 16) return;           // uniform per wave -> EXEC all-1s
    int l  = lane & 15;
    int hi = lane >> 4;
    int row0 = wave * 16;
    const float* xrow = x + (size_t)(row0 + l) * DIN;

    v8f acc0 = {}, acc1 = {}, acc2 = {}, acc3 = {};
    for (int k0 = 0; k0 < DIN; k0 += 4) {
        int ka = k0 + 2 * hi;                   // this lane's K pair
        v2f a = *(const v2f*)(xrow + ka);       // A[row][ka], A[row][ka+1]
        const float* w0r = W0 + (size_t)ka * DLAT;
        v2f b0t = { w0r[ 0 + l], w0r[DLAT +  0 + l] };
        v2f b1t = { w0r[16 + l], w0r[DLAT + 16 + l] };
        v2f b2t = { w0r[32 + l], w0r[DLAT + 32 + l] };
        v2f b3t = { w0r[48 + l], w0r[DLAT + 48 + l] };
        acc0 = __builtin_amdgcn_wmma_f32_16x16x4_f32(false, a, false, b0t, (short)0, acc0, false, false);
        acc1 = __builtin_amdgcn_wmma_f32_16x16x4_f32(false, a, false, b1t, (short)0, acc1, false, false);
        acc2 = __builtin_amdgcn_wmma_f32_16x16x4_f32(false, a, false, b2t, (short)0, acc2, false, false);
        acc3 = __builtin_amdgcn_wmma_f32_16x16x4_f32(false, a, false, b3t, (short)0, acc3, false, false);
    }
    // D layout: VGPR r -> rows r (lanes 0-15) / r+8 (lanes 16-31), col = l + tile*16
    for (int r = 0; r < 8; ++r) {
        int orow = row0 + r + 8 * hi;
        float* hrow = h0 + (size_t)orow * DLAT;
        hrow[ 0 + l] = acc0[r] + b0[ 0 + l];
        hrow[16 + l] = acc1[r] + b0[16 + l];
        hrow[32 + l] = acc2[r] + b0[32 + l];
        hrow[48 + l] = acc3[r] + b0[48 + l];
    }
}

// -------- per-node attention scalars: ai = h0 . Watt[:64], aj = h0 . Watt[64:] --------
__global__ void gat_att_scalars_kernel(const float* __restrict__ h0,
                                       const float* __restrict__ Watt,
                                       float* __restrict__ ai,
                                       float* __restrict__ aj) {
    int n = blockIdx.x * blockDim.x + threadIdx.x;
    if (n >= N_NODES) return;
    const float* hn = h0 + (size_t)n * DLAT;
    float ai0 = 0.f, ai1 = 0.f, aj0 = 0.f, aj1 = 0.f;
    for (int d = 0; d < DLAT; ++d) {
        float v = hn[d];
        ai0 += v * Watt[d * NHEAD + 0];
        ai1 += v * Watt[d * NHEAD + 1];
        aj0 += v * Watt[(DLAT + d) * NHEAD + 0];
        aj1 += v * Watt[(DLAT + d) * NHEAD + 1];
    }
    ai[n * 2 + 0] = ai0;  ai[n * 2 + 1] = ai1;
    aj[n * 2 + 0] = aj0;  aj[n * 2 + 1] = aj1;
}

__device__ __forceinline__ void atomicMaxF32(float* addr, float val) {
    if (val >= 0.0f) atomicMax((int*)addr, __float_as_int(val));
    else             atomicMin((unsigned int*)addr, (unsigned int)__float_as_int(val));
}

// -------- edge logits + segment max --------
__global__ void gat_edge_logit_kernel(const int* __restrict__ src, const int* __restrict__ dst,
                                      const float* __restrict__ ai, const float* __restrict__ aj,
                                      float* __restrict__ elog, float* __restrict__ emax) {
    int e = blockIdx.x * blockDim.x + threadIdx.x;
    if (e >= N_EDGES) return;
    int s = src[e], d = dst[e];
    float e0 = ai[d * 2 + 0] + aj[s * 2 + 0];
    float e1 = ai[d * 2 + 1] + aj[s * 2 + 1];
    e0 = (e0 > 0.f) ? e0 : 0.2f * e0;   // leaky_relu(0.2)
    e1 = (e1 > 0.f) ? e1 : 0.2f * e1;
    elog[e * 2 + 0] = e0;
    elog[e * 2 + 1] = e1;
    atomicMaxF32(emax + d * 2 + 0, e0);
    atomicMaxF32(emax + d * 2 + 1, e1);
}

// -------- edge exp + segment sum (ee overwrites elog) --------
__global__ void gat_edge_exp_kernel(const int* __restrict__ dst,
                                    float* __restrict__ elog,
                                    const float* __restrict__ emax,
                                    float* __restrict__ denom) {
    int e = blockIdx.x * blockDim.x + threadIdx.x;
    if (e >= N_EDGES) return;
    int d = dst[e];
    float ee0 = __expf(elog[e * 2 + 0] - emax[d * 2 + 0]);
    float ee1 = __expf(elog[e * 2 + 1] - emax[d * 2 + 1]);
    elog[e * 2 + 0] = ee0;
    elog[e * 2 + 1] = ee1;
    atomicAdd(denom + d * 2 + 0, ee0);
    atomicAdd(denom + d * 2 + 1, ee1);
}

// -------- weighted message scatter: one wave32 per edge, float2 per lane --------
__global__ void gat_message_kernel(const int* __restrict__ src, const int* __restrict__ dst,
                                   const float* __restrict__ h0,
                                   const float* __restrict__ ee,
                                   const float* __restrict__ denom,
                                   float* __restrict__ hagg) {
    int gid  = blockIdx.x * blockDim.x + threadIdx.x;
    int e    = gid >> 5;
    int lane = gid & 31;
    if (e >= N_EDGES) return;
    int s = src[e], d = dst[e];
    float a0 = ee[e * 2 + 0] / denom[d * 2 + 0];
    float a1 = ee[e * 2 + 1] / denom[d * 2 + 1];
    float w  = 0.5f * (a0 + a1);                 // head-mean folded in
    v2f hv = *(const v2f*)(h0 + (size_t)s * DLAT + 2 * lane);
    atomicAdd(hagg + (size_t)d * DLAT + 2 * lane + 0, w * hv.x);
    atomicAdd(hagg + (size_t)d * DLAT + 2 * lane + 1, w * hv.y);
}

// -------- global mean pool: one wave32 per node --------
__global__ void gat_pool_kernel(const int* __restrict__ batch,
                                const float* __restrict__ hagg,
                                float* __restrict__ gsum, float* __restrict__ cnt) {
    int gid  = blockIdx.x * blockDim.x + threadIdx.x;
    int n    = gid >> 5;
    int lane = gid & 31;
    if (n >= N_NODES) return;
    int g = batch[n];
    v2f hv = *(const v2f*)(hagg + (size_t)n * DLAT + 2 * lane);
    atomicAdd(gsum + (size_t)g * DLAT + 2 * lane + 0, hv.x);
    atomicAdd(gsum + (size_t)g * DLAT + 2 * lane + 1, hv.y);
    if (lane == 0) atomicAdd(cnt + g, 1.0f);
}

// -------- output head: (gsum/max(cnt,1)) @ Wout + bout --------
__global__ void gat_out_kernel(const float* __restrict__ gsum, const float* __restrict__ cnt,
                               const float* __restrict__ Wout, const float* __restrict__ bout,
                               float* __restrict__ out) {
    int t = blockIdx.x * blockDim.x + threadIdx.x;
    if (t >= NUM_GRAPHS * NCLS) return;
    int g = t / NCLS, c = t % NCLS;
    float cg = cnt[g];
    cg = (cg > 1.0f) ? cg : 1.0f;
    float s = 0.f;
    for (int k = 0; k < DLAT; ++k)
        s += (gsum[g * DLAT + k] / cg) * Wout[k * NCLS + c];
    out[t] = s + bout[c];
}

extern "C" void kernel_launch(void* const* d_in, const int* in_sizes, int n_in,
                              void* d_out, int out_size, void* d_ws, size_t ws_size,
                              hipStream_t stream) {
    const float* x     = (const float*)d_in[0];   // [N,128]
    const int*   eidx  = (const int*)  d_in[1];   // [2,E]
    const int*   batch = (const int*)  d_in[2];   // [N]
    const float* W0    = (const float*)d_in[3];   // [128,64]
    const float* b0    = (const float*)d_in[4];   // [64]
    const float* Watt  = (const float*)d_in[5];   // [128,2]
    const float* Wout  = (const float*)d_in[6];   // [64,10]
    const float* bout  = (const float*)d_in[7];   // [10]
    float*       out   = (float*)d_out;           // [64,10]

    const int* src = eidx;            // row 0
    const int* dst = eidx + N_EDGES;  // row 1

    float* ws   = (float*)d_ws;
    float* h0   = ws + OFF_H0;
    float* ai   = ws + OFF_AI;
    float* aj   = ws + OFF_AJ;
    float* emax = ws + OFF_EMAX;
    float* den  = ws + OFF_DEN;
    float* ee   = ws + OFF_EE;
    float* hagg = ws + OFF_HAGG;
    float* gsum = ws + OFF_GSUM;
    float* cnt  = ws + OFF_CNT;

    // 1) init workspace
    gat_init_kernel<<<2048, 256, 0, stream>>>(ws);

    // 2) h0 = x @ W0 + b0   (6250 waves of 16x64 tiles, 8 waves/block)
    {
        int waves = N_NODES / 16;                 // 6250, exact
        int blocks = (waves + 7) / 8;             // 8 waves per 256-thread block
        gat_gemm_h0_kernel<<<blocks, 256, 0, stream>>>(x, W0, b0, h0);
    }

    // 3) per-node attention scalars
    gat_att_scalars_kernel<<<(N_NODES + 255) / 256, 256, 0, stream>>>(h0, Watt, ai, aj);

    // 4) edge logits + segment max
    gat_edge_logit_kernel<<<(N_EDGES + 255) / 256, 256, 0, stream>>>(src, dst, ai, aj, ee, emax);

    // 5) edge exp + segment sum
    gat_edge_exp_kernel<<<(N_EDGES + 255) / 256, 256, 0, stream>>>(dst, ee, emax, den);

    // 6) weighted message scatter (wave per edge)
    {
        long long threads = (long long)N_EDGES * 32;
        int blocks = (int)((threads + 255) / 256);
        gat_message_kernel<<<blocks, 256, 0, stream>>>(src, dst, h0, ee, den, hagg);
    }

    // 7) global mean pool (wave per node)
    {
        long long threads = (long long)N_NODES * 32;
        int blocks = (int)((threads + 255) / 256);
        gat_pool_kernel<<<blocks, 256, 0, stream>>>(batch, hagg, gsum, cnt);
    }

    // 8) output head
    gat_out_kernel<<<(NUM_GRAPHS * NCLS + 255) / 256, 256, 0, stream>>>(gsum, cnt, Wout, bout, out);
}